// VoxelHashTable_738734375104
// MI455X (gfx1250) — compile-verified
//
#include <hip/hip_runtime.h>
#include <hip/hip_bf16.h>
#include <math.h>

// ---------------------------------------------------------------------------
// VoxelHashTable gather: out[q, :] = voxel_features[table[hash(q)], :] or 0
// Pure HBM-bandwidth-bound row gather (~1.6 GB traffic -> ~66us floor at
// 23.3 TB/s). No matmul -> WMMA inapplicable. We use the CDNA5 async
// global<->LDS copy path (ASYNCcnt) so row copies never touch VGPRs.
// Hash is computed in 32-bit: mod 2^20 == mask of low 20 bits, and 32-bit
// two's-complement wraparound preserves those bits exactly.
// ---------------------------------------------------------------------------

#define FDIM            768
#define VEC4_PER_ROW    (FDIM / 4)             // 192 float4 per row
#define LANES           32
#define V4_PER_LANE     (VEC4_PER_ROW / LANES) // 6 float4 per lane
#define WAVES_PER_BLOCK 8
#define BLOCK_THREADS   (WAVES_PER_BLOCK * LANES)
#define TABLE_MASK      0xFFFFFu               // TABLE_SIZE = 2^20

#if defined(__HIP_DEVICE_COMPILE__) && \
    __has_builtin(__builtin_amdgcn_global_load_async_to_lds_b128) && \
    __has_builtin(__builtin_amdgcn_global_store_async_from_lds_b128)
#define USE_ASYNC_LDS 1
#else
#define USE_ASYNC_LDS 0
#endif

#if USE_ASYNC_LDS
// Builtin signature (from clang diagnostic): 16-byte int vector pointers,
// global AS for the memory side, LDS AS for the LDS side.
typedef int v4i __attribute__((ext_vector_type(4)));
typedef __attribute__((address_space(1))) v4i gv4i;
typedef __attribute__((address_space(3))) v4i lv4i;

__device__ __forceinline__ void async_row_in(const float* g, float* l) {
    // per-lane 16B: global -> LDS, tracked by ASYNCcnt
    __builtin_amdgcn_global_load_async_to_lds_b128((gv4i*)g, (lv4i*)l, 0, 0);
}
__device__ __forceinline__ void async_row_out(float* g, const float* l) {
    // per-lane 16B: LDS -> global, tracked by ASYNCcnt
    __builtin_amdgcn_global_store_async_from_lds_b128((gv4i*)g, (lv4i*)l, 0, 0);
}
__device__ __forceinline__ void wait_async0() {
#if __has_builtin(__builtin_amdgcn_s_wait_asynccnt)
    __builtin_amdgcn_s_wait_asynccnt(0);
#else
    asm volatile("s_wait_asynccnt 0" ::: "memory");
#endif
}
#endif

__global__ __launch_bounds__(BLOCK_THREADS)
void voxel_hash_gather_kernel(const float* __restrict__ qpts,
                              const float* __restrict__ feats,
                              const int*   __restrict__ table,
                              float*       __restrict__ out,
                              int M)
{
    __shared__ float smem[WAVES_PER_BLOCK * FDIM];   // 24 KB / block

    const int lane = threadIdx.x & (LANES - 1);
    const int wave = threadIdx.x >> 5;
    float* wbuf = &smem[wave * FDIM];

    int q = blockIdx.x * WAVES_PER_BLOCK + wave;
    const int qstride = gridDim.x * WAVES_PER_BLOCK;

    for (; q < M; q += qstride) {
        // ---- hash (wave-uniform; redundant per lane, L1-broadcast) --------
        const float px = qpts[3 * q + 0];
        const float py = qpts[3 * q + 1];
        const float pz = qpts[3 * q + 2];
        // mirrors jnp.floor(p / 0.1f) in float32 exactly; grid coords are
        // tiny (|g| < 256) so int32 holds them, and the mod-2^20 hash only
        // needs the low 20 bits -> unsigned 32-bit wraparound arithmetic.
        const int gx = (int)floorf(px / 0.1f);
        const int gy = (int)floorf(py / 0.1f);
        const int gz = (int)floorf(pz / 0.1f);
        const unsigned hs = (unsigned)gx * 73856093u
                          + (unsigned)gy * 19349669u
                          + (unsigned)gz * 83492791u;
        const int vidx = table[hs & TABLE_MASK];

        float* orow = out + (size_t)q * FDIM;

        if (vidx < 0) {
            // invalid voxel: write zeros, 16B per lane per iter (512B/wave/instr)
            const float4 z = make_float4(0.f, 0.f, 0.f, 0.f);
            #pragma unroll
            for (int i = 0; i < V4_PER_LANE; ++i)
                ((float4*)orow)[lane + i * LANES] = z;
        } else {
            const float* frow = feats + (size_t)vidx * FDIM;
#if USE_ASYNC_LDS
            // ---- CDNA5 async path: HBM -> LDS -> HBM, zero VGPR traffic ---
            #pragma unroll
            for (int i = 0; i < V4_PER_LANE; ++i) {
                const int off = (lane + i * LANES) * 4;     // float index
                async_row_in(frow + off, wbuf + off);
            }
            wait_async0();                                  // row resident in LDS
            #pragma unroll
            for (int i = 0; i < V4_PER_LANE; ++i) {
                const int off = (lane + i * LANES) * 4;
                async_row_out(orow + off, wbuf + off);
            }
            wait_async0();                                  // LDS buffer reusable
#else
            // ---- fallback: wide VGPR copy (b128 loads/stores) -------------
            float4 r[V4_PER_LANE];
            #pragma unroll
            for (int i = 0; i < V4_PER_LANE; ++i)
                r[i] = ((const float4*)frow)[lane + i * LANES];
            #pragma unroll
            for (int i = 0; i < V4_PER_LANE; ++i)
                ((float4*)orow)[lane + i * LANES] = r[i];
#endif
        }
    }
}

extern "C" void kernel_launch(void* const* d_in, const int* in_sizes, int n_in,
                              void* d_out, int out_size, void* d_ws, size_t ws_size,
                              hipStream_t stream) {
    const float* qpts  = (const float*)d_in[0];   // (M, 3) float32
    const float* feats = (const float*)d_in[1];   // (V, 768) float32
    const int*   table = (const int*)d_in[2];     // (2^20,) int (vidx fits in i32)
    float*       out   = (float*)d_out;           // (M, 768) float32

    const int M = in_sizes[0] / 3;
    const int blocks = (M + WAVES_PER_BLOCK - 1) / WAVES_PER_BLOCK;

    voxel_hash_gather_kernel<<<blocks, BLOCK_THREADS, 0, stream>>>(
        qpts, feats, table, out, M);
}